// TemporalMemory_55533927138024
// MI455X (gfx1250) — compile-verified
//
#include <hip/hip_runtime.h>
#include <hip/hip_bf16.h>

// ---------------------------------------------------------------------------
// TemporalMemory step for MI455X (gfx1250, wave32).
//
// Heavy op: prediction[j] = sum_{i in active cells} W[i][j]  (binary mask).
// Only rows inside the 128 active columns (<=1024 of 16384) are read: <=64 MB
// of HBM traffic instead of the dense 1 GB -> memory-bound, ~2.8us floor.
//
// tm_select compacts the live 4-row chunks (rowbase + float4 mask); the
// matvec then runs a branch-free, software-pipelined loop of
// V_WMMA_F32_16X16X4_F32 ops: A(16x4) = mask replicated over M,
// B(4x16) = W tile. Each wave owns 32 output columns (2 WMMA tiles).
// ---------------------------------------------------------------------------

#define TM_COLS    2048
#define TM_CELLS   8
#define TM_N       16384      // TM_COLS * TM_CELLS
#define TM_NACT    128
#define TM_MAXCH   (TM_NACT * 2)
#define TM_EPS     1e-8f
#define TM_THRESH  0.5f
#define PF_DIST    6

typedef float v2f __attribute__((ext_vector_type(2)));
typedef float v8f __attribute__((ext_vector_type(8)));

// d_out layout (floats):
//   [0            , TM_N)            cell_activation
//   [TM_N         , 2*TM_N)          winner_cells
//   [2*TM_N       , 3*TM_N)          cell_prediction
//   [3*TM_N       , 3*TM_N+TM_NACT)  active_column_bursting
//   [3*TM_N+NACT  , ...+TM_N)        prediction (float)
//
// d_ws layout (bytes):
//   [0   , 4)      nchunks (int)
//   [16  , 1040)   chunk row base, int[256]
//   [1040, 5136)   chunk mask, float4[256]

__global__ void tm_zero(float* __restrict__ out, int n) {
    int i = blockIdx.x * blockDim.x + threadIdx.x;
    for (; i < n; i += gridDim.x * blockDim.x) out[i] = 0.0f;
}

__global__ void tm_select(const int*   __restrict__ active_columns,
                          const int*   __restrict__ prev_pred,    // bool-as-int
                          const float* __restrict__ prev_pot,
                          const int*   __restrict__ bundle_seg,
                          const float* __restrict__ lu_jitter,
                          float* __restrict__ out,
                          int*    __restrict__ ws_n,
                          int*    __restrict__ ws_row,
                          float4* __restrict__ ws_mask)
{
    __shared__ float s_m[TM_NACT * TM_CELLS];
    __shared__ int   s_rb[TM_NACT];

    const int a = threadIdx.x;
    if (a < TM_NACT) {
        const int col  = active_columns[a];
        const int base = col * TM_CELLS;
        // active_columns is sorted; dedupe repeats (count each column once).
        const bool dup = (a > 0) && (active_columns[a - 1] == col);

        float acp[TM_CELLS], pot[TM_CELLS], seg[TM_CELLS];
        bool  anyp = false;
        float mx = -1e30f, mn = 1e30f;
        #pragma unroll
        for (int k = 0; k < TM_CELLS; ++k) {
            const bool p = prev_pred[base + k] != 0;
            acp[k] = p ? 1.0f : 0.0f;
            anyp  |= p;
            pot[k] = prev_pot[base + k];
            mx = fmaxf(mx, pot[k]);
            seg[k] = (float)bundle_seg[base + k] + lu_jitter[a * TM_CELLS + k];
            mn = fminf(mn, seg[k]);
        }
        const bool bursting = !anyp;
        const bool matching = mx >= TM_THRESH;

        out[3 * TM_N + a] = bursting ? 1.0f : 0.0f;   // per-a, always written
        s_rb[a] = base;

        #pragma unroll
        for (int k = 0; k < TM_CELLS; ++k) {
            const float best   = (fabsf(pot[k] - mx) < TM_EPS) ? 1.0f : 0.0f;
            const float least  = (fabsf(seg[k] - mn) < TM_EPS) ? 1.0f : 0.0f;
            const float winner = (acp[k] != 0.0f) ? 1.0f
                                : (bursting ? (matching ? best : least) : 0.0f);
            const float actv   = (acp[k] != 0.0f || bursting) ? 1.0f : 0.0f;
            s_m[a * TM_CELLS + k] = dup ? 0.0f : actv;
            if (!dup) {
                out[base + k]        = actv;    // cell_activation
                out[TM_N + base + k] = winner;  // winner_cells
            }
        }
    }
    __syncthreads();

    // Deterministic serial compaction of live 4-row chunks (256 items).
    if (threadIdx.x == 0) {
        int cnt = 0;
        for (int q = 0; q < TM_MAXCH; ++q) {
            const int a2 = q >> 1, ch = q & 1;
            const float m0 = s_m[a2 * TM_CELLS + ch * 4 + 0];
            const float m1 = s_m[a2 * TM_CELLS + ch * 4 + 1];
            const float m2 = s_m[a2 * TM_CELLS + ch * 4 + 2];
            const float m3 = s_m[a2 * TM_CELLS + ch * 4 + 3];
            if (m0 + m1 + m2 + m3 != 0.0f) {
                ws_row[cnt] = s_rb[a2] + ch * 4;
                ws_mask[cnt] = make_float4(m0, m1, m2, m3);
                ++cnt;
            }
        }
        ws_n[0] = cnt;
    }
}

// 64 blocks x 256 threads (8 waves); wave owns 32 output columns (2 tiles).
// Branch-free pipelined loop: loads for chunk c+1 issue before the WMMAs of
// chunk c, overlapping fetch latency with mask handling + 2 XDL ops.
__global__ void __launch_bounds__(256)
tm_matvec_wmma(const float*  __restrict__ W,
               const int*    __restrict__ ws_n,
               const int*    __restrict__ ws_row,
               const float4* __restrict__ ws_mask,
               float* __restrict__ out)
{
    __shared__ int    s_row[TM_MAXCH];
    __shared__ float4 s_mask[TM_MAXCH];
    __shared__ int    s_n;

    if (threadIdx.x == 0) s_n = ws_n[0];
    for (int i = threadIdx.x; i < TM_MAXCH; i += blockDim.x) {
        s_row[i]  = ws_row[i];
        s_mask[i] = ws_mask[i];
    }
    __syncthreads();
    const int nch = s_n;

    const int lane = threadIdx.x & 31;
    const int wave = threadIdx.x >> 5;
    const int half = lane >> 4;          // B rows K0/K1 (half=0) vs K2/K3
    const int n    = lane & 15;          // column within a 16-wide tile
    const int j0   = (blockIdx.x * 8 + wave) * 32;
    const size_t off0 = (size_t)(j0 + n) + (size_t)(half * 2) * (size_t)TM_N;

    v8f acc0 = {0.f, 0.f, 0.f, 0.f, 0.f, 0.f, 0.f, 0.f};
    v8f acc1 = {0.f, 0.f, 0.f, 0.f, 0.f, 0.f, 0.f, 0.f};

    if (nch > 0) {
        const float* p = W + (((size_t)s_row[0]) << 14) + off0;
        float b0x = p[0], b0y = p[TM_N], b1x = p[16], b1y = p[16 + TM_N];

        for (int c = 0; c < nch; ++c) {
            // Issue next chunk's B loads first (index clamped: branch-free).
            const int cn = (c + 1 < nch) ? (c + 1) : c;
            const float* pn = W + (((size_t)s_row[cn]) << 14) + off0;
            const float nb0x = pn[0],  nb0y = pn[TM_N];
            const float nb1x = pn[16], nb1y = pn[16 + TM_N];

            if (c + PF_DIST < nch)
                __builtin_prefetch(
                    W + (((size_t)s_row[c + PF_DIST]) << 14) + off0, 0, 0);

            // A 16x4 f32, mask replicated over M (ISA 7.12.2 layout):
            // VGPR0 = K0 (lanes0-15)/K2 (lanes16-31), VGPR1 = K1/K3.
            const float4 m = s_mask[c];
            v2f av; av.x = half ? m.z : m.x; av.y = half ? m.w : m.y;
            v2f bv0; bv0.x = b0x; bv0.y = b0y;
            v2f bv1; bv1.x = b1x; bv1.y = b1y;

            acc0 = __builtin_amdgcn_wmma_f32_16x16x4_f32(
                false, av, false, bv0, (short)0, acc0, false, false);
            acc1 = __builtin_amdgcn_wmma_f32_16x16x4_f32(
                false, av, false, bv1, (short)0, acc1, false, false);

            b0x = nb0x; b0y = nb0y; b1x = nb1x; b1y = nb1y;
        }
    }

    // D layout: VGPR0 = row M=0 (lanes0-15)/M=8 (lanes16-31); A replicated
    // over M makes all D rows equal, so lanes 0-15 carry all 16 columns.
    if (lane < 16) {
        const int   j  = j0 + n;
        const float r0 = acc0[0];
        const float r1 = acc1[0];
        out[3 * TM_N + TM_NACT + j]      = r0;                       // prediction
        out[3 * TM_N + TM_NACT + j + 16] = r1;
        out[2 * TM_N + j]      = (r0 > TM_EPS) ? 1.0f : 0.0f;        // cell_pred
        out[2 * TM_N + j + 16] = (r1 > TM_EPS) ? 1.0f : 0.0f;
    }
}

extern "C" void kernel_launch(void* const* d_in, const int* in_sizes, int n_in,
                              void* d_out, int out_size, void* d_ws, size_t ws_size,
                              hipStream_t stream) {
    (void)in_sizes; (void)n_in; (void)out_size; (void)ws_size;

    const int*   active_columns = (const int*)  d_in[0];
    const int*   prev_pred      = (const int*)  d_in[1];  // bool-as-int
    const float* prev_pot       = (const float*)d_in[2];
    const int*   bundle_seg     = (const int*)  d_in[3];
    const float* lu_jitter      = (const float*)d_in[4];
    const float* W              = (const float*)d_in[5];

    float*  out     = (float*)d_out;
    int*    ws_nch  = (int*)d_ws;
    int*    ws_row  = (int*)((char*)d_ws + 16);
    float4* ws_mask = (float4*)((char*)d_ws + 16 + TM_MAXCH * sizeof(int));

    // 1) zero the scatter targets (cell_activation, winner_cells)
    tm_zero<<<128, 256, 0, stream>>>(out, 2 * TM_N);

    // 2) winner/activation selection + live-chunk compaction
    tm_select<<<1, TM_NACT, 0, stream>>>(active_columns, prev_pred, prev_pot,
                                         bundle_seg, lu_jitter, out,
                                         ws_nch, ws_row, ws_mask);

    // 3) masked row-sum over W via pipelined WMMA f32 16x16x4
    tm_matvec_wmma<<<TM_N / 256, 256, 0, stream>>>(W, ws_nch, ws_row, ws_mask,
                                                   out);
}